// GNNModel_68298569941218
// MI455X (gfx1250) — compile-verified
//
#include <hip/hip_runtime.h>

// ---------------------------------------------------------------------------
// 2-layer GCN for MI455X (gfx1250, wave32).
//   deg/dis -> edge prep (int32 idx + fused symmetric norm, reused 2x) ->
//   [X@W1: V_WMMA_F32_16X16X4_F32, W staged in LDS via async global->LDS] ->
//   selfloop+bias init -> edge scatter (f32 atomics, L2-resident) ->
//   [ReLU-on-load GEMM2 via WMMA] -> init into d_out -> scatter into d_out.
// ---------------------------------------------------------------------------

typedef __attribute__((ext_vector_type(2))) float v2f;
typedef __attribute__((ext_vector_type(8))) float v8f;
typedef __attribute__((ext_vector_type(4))) int   v4i;

#define GCN_C_IN  128
#define GCN_C_HID 64

#if defined(__AMDGCN__) && \
    __has_builtin(__builtin_amdgcn_global_load_async_to_lds_b128) && \
    __has_builtin(__builtin_amdgcn_s_wait_asynccnt)
#define GCN_ASYNC_LDS 1
#else
#define GCN_ASYNC_LDS 0
#endif

// ---- degree / normalization ------------------------------------------------
__global__ void gcn_deg_init(float* __restrict__ deg, int n) {
  int i = blockIdx.x * blockDim.x + threadIdx.x;
  if (i < n) deg[i] = 1.0f;  // self-loop
}

__global__ void gcn_deg_count(const long long* __restrict__ dst,
                              float* __restrict__ deg, int e) {
  int i = blockIdx.x * blockDim.x + threadIdx.x;
  if (i < e) unsafeAtomicAdd(&deg[(int)dst[i]], 1.0f);
}

__global__ void gcn_rsqrt(const float* __restrict__ deg,
                          float* __restrict__ dis, int n) {
  int i = blockIdx.x * blockDim.x + threadIdx.x;
  if (i < n) dis[i] = __frsqrt_rn(deg[i]);
}

// ---- edge prep: int64 -> int32 indices + per-edge symmetric norm -----------
__global__ void gcn_edge_prep(const long long* __restrict__ src64,
                              const long long* __restrict__ dst64,
                              const float* __restrict__ dis,
                              int* __restrict__ s32, int* __restrict__ d32,
                              float* __restrict__ norm, int e) {
  int i = blockIdx.x * blockDim.x + threadIdx.x;
  if (i >= e) return;
  int s = (int)src64[i];
  int d = (int)dst64[i];
  s32[i] = s;
  d32[i] = d;
  norm[i] = dis[s] * dis[d];
}

// ---- WMMA GEMM: out[M x 64] = act(A[M x K]) @ W[K x 64], fp32 --------------
// Block = 128 threads (4 waves). The block stages all of W into LDS (async
// global->LDS on gfx1250), then each wave computes one 16x16 tile of the
// block's 16-row M stripe (wave w -> N tile w). A-fragment per-lane layout for
// V_WMMA_F32_16X16X4_F32 (ISA 7.12.2): lanes 0-15 hold K={k,k+1} of row
// M=lane, lanes 16-31 hold K={k+2,k+3} of row M=lane-16; B mirrors the K
// split. C/D: VGPR r = row (r + 8*half), col = lane&15.
template <int RELU, int K>
__global__ void gcn_gemm_wmma(const float* __restrict__ A,
                              const float* __restrict__ W,
                              float* __restrict__ out, int M) {
  __shared__ float ldsW[K * 64];
  constexpr int TOT = K * 64;                     // 8192 (K=128) / 4096 (K=64)
  const int tid = threadIdx.x;

#if GCN_ASYNC_LDS
  typedef __attribute__((address_space(1))) v4i GV4i;
  typedef __attribute__((address_space(3))) v4i LV4i;
#pragma unroll
  for (int off = tid * 4; off < TOT; off += 128 * 4) {
    __builtin_amdgcn_global_load_async_to_lds_b128(
        (GV4i*)(W + off), (LV4i*)&ldsW[off], /*offset=*/0, /*cpol=*/0);
  }
  __builtin_amdgcn_s_wait_asynccnt(0);
  __syncthreads();
#else
#pragma unroll
  for (int off = tid * 4; off < TOT; off += 128 * 4) {
    *(float4*)&ldsW[off] = *(const float4*)(W + off);
  }
  __syncthreads();
#endif

  const int lane  = tid & 31;
  const int wave  = tid >> 5;                     // 0..3 -> N tile
  const int Mbase = blockIdx.x << 4;
  const int Nbase = wave << 4;
  const int half  = lane >> 4;                    // 0: K lo pair, 1: K hi pair
  const int l15   = lane & 15;
  int rowA = Mbase + l15;
  if (rowA >= M) rowA = M - 1;                    // clamp loads; stores guarded
  const float* __restrict__ arow = A + (size_t)rowA * K;

  v8f acc = {};
#pragma unroll 4
  for (int k = 0; k < K; k += 4) {
    const int ka = k + 2 * half;
    v2f a = *(const v2f*)(arow + ka);             // 8B-aligned (ka even)
    if (RELU) { a.x = fmaxf(a.x, 0.0f); a.y = fmaxf(a.y, 0.0f); }
    v2f b;
    b.x = ldsW[ka * 64 + Nbase + l15];
    b.y = ldsW[(ka + 1) * 64 + Nbase + l15];
    acc = __builtin_amdgcn_wmma_f32_16x16x4_f32(
        /*neg_a=*/false, a, /*neg_b=*/false, b,
        /*c_mod=*/(short)0, acc, /*reuse_a=*/false, /*reuse_b=*/false);
  }
#pragma unroll
  for (int r = 0; r < 8; ++r) {
    const int row = Mbase + r + 8 * half;
    if (row < M) out[(size_t)row * 64 + Nbase + l15] = acc[r];
  }
}

// ---- self-loop + bias init: out[i,c] = dis[i]^2 * h[i,c] + b[c] ------------
__global__ void gcn_self_init(const float* __restrict__ dis,
                              const float* __restrict__ h,
                              const float* __restrict__ bias,
                              float* __restrict__ out, int total) {
  int t = blockIdx.x * blockDim.x + threadIdx.x;
  if (t >= total) return;
  int i = t >> 6;                                 // C = 64
  int c = t & 63;
  float di = dis[i];
  out[t] = di * di * h[t] + bias[c];
}

// ---- edge scatter: out[dst] += norm[e]*h[src], one wave per edge -----------
__global__ void gcn_edge_scatter(const int* __restrict__ s32,
                                 const int* __restrict__ d32,
                                 const float* __restrict__ norm,
                                 const float* __restrict__ h,
                                 float* __restrict__ out, int e) {
  int gid  = blockIdx.x * blockDim.x + threadIdx.x;
  int edge = gid >> 5;
  if (edge >= e) return;
  int lane = gid & 31;
  int s = s32[edge];
  int d = d32[edge];
  float w = norm[edge];
  int c = lane << 1;                              // 2 channels/lane (64 total)
  const float2 hv = *(const float2*)(h + (size_t)s * 64 + c);
  float* o = out + (size_t)d * 64 + c;
  unsafeAtomicAdd(o,     w * hv.x);
  unsafeAtomicAdd(o + 1, w * hv.y);
}

// ---------------------------------------------------------------------------
extern "C" void kernel_launch(void* const* d_in, const int* in_sizes, int n_in,
                              void* d_out, int out_size, void* d_ws, size_t ws_size,
                              hipStream_t stream) {
  const float*     x   = (const float*)d_in[0];
  const long long* ei  = (const long long*)d_in[1];   // int64 [2, E]
  const float*     W1  = (const float*)d_in[2];
  const float*     b1  = (const float*)d_in[3];
  const float*     W2  = (const float*)d_in[4];
  const float*     b2  = (const float*)d_in[5];
  float*           out = (float*)d_out;

  const int N = in_sizes[0] / GCN_C_IN;   // 50000
  const int E = in_sizes[1] / 2;          // 800000
  const long long* src64 = ei;            // row 0
  const long long* dst64 = ei + E;        // row 1

  // workspace (floats/ints, 4B each):
  // deg[N] | dis[N] | h1[N*64] | agg1[N*64] | h2[N*64] | s32[E] | d32[E] | norm[E]
  float* ws   = (float*)d_ws;
  float* deg  = ws;
  float* dis  = ws + N;
  float* h1   = ws + 2 * (size_t)N;
  float* agg1 = h1 + (size_t)N * 64;
  float* h2   = agg1 + (size_t)N * 64;
  int*   s32  = (int*)(h2 + (size_t)N * 64);
  int*   d32  = s32 + E;
  float* norm = (float*)(d32 + E);

  const int T = 256;
  const int total64 = N * 64;
  const int mblocks = (N + 15) / 16;                       // one block per M stripe
  const int sblocks = (int)(((size_t)E * 32 + T - 1) / T); // scatter grid

  // degree + symmetric norm, then per-edge packed metadata (reused by both layers)
  gcn_deg_init <<<(N + T - 1) / T, T, 0, stream>>>(deg, N);
  gcn_deg_count<<<(E + T - 1) / T, T, 0, stream>>>(dst64, deg, E);
  gcn_rsqrt    <<<(N + T - 1) / T, T, 0, stream>>>(deg, dis, N);
  gcn_edge_prep<<<(E + T - 1) / T, T, 0, stream>>>(src64, dst64, dis, s32, d32, norm, E);

  // layer 1: h1 = x @ W1 ; agg1 = dis^2*h1 + b1 ; scatter edges into agg1
  gcn_gemm_wmma<0, GCN_C_IN><<<mblocks, 128, 0, stream>>>(x, W1, h1, N);
  gcn_self_init<<<(total64 + T - 1) / T, T, 0, stream>>>(dis, h1, b1, agg1, total64);
  gcn_edge_scatter<<<sblocks, T, 0, stream>>>(s32, d32, norm, h1, agg1, E);

  // layer 2: h2 = relu(agg1) @ W2 (ReLU fused into A-fragment load)
  gcn_gemm_wmma<1, GCN_C_HID><<<mblocks, 128, 0, stream>>>(agg1, W2, h2, N);
  gcn_self_init<<<(total64 + T - 1) / T, T, 0, stream>>>(dis, h2, b2, out, total64);
  gcn_edge_scatter<<<sblocks, T, 0, stream>>>(s32, d32, norm, h2, out, E);
}